// Model_41266045780663
// MI455X (gfx1250) — compile-verified
//
#include <hip/hip_runtime.h>
#include <math.h>

typedef __attribute__((ext_vector_type(2))) float v2f;
typedef __attribute__((ext_vector_type(8))) float v8f;

#define TSEQ   2048
#define TP     2053      // 2 + 2048 + 3
#define PADL   2
#define IDIM   32
#define HDIM   64
#define ODIM   16
// pair-interleaved hT layout: addr(row,col) = (row>>1)*PSTR + 2*col + (row&1)
#define PSTR   32        // dwords per row-pair (16 cols * 2)
#define HREG   ((HDIM/2) * PSTR)   // 1024 dwords per buffer

// Hardware tanh (TRANS pipe, co-issues with the matrix pipe).
__device__ __forceinline__ float hw_tanh(float v) {
#if __has_builtin(__builtin_amdgcn_tanhf)
    return __builtin_amdgcn_tanhf(v);
#else
    float r;
    asm volatile("v_tanh_f32 %0, %1\n\tv_nop" : "=v"(r) : "v"(v));
    return r;
#endif
}

// 16 blocks x 128 threads. Each block owns a 16-row batch tile; each of its
// 4 waves owns one 16-row slice of H (one WMMA D tile). hT formulation:
// D = W_hh x hT + W_ih x xT + bias, h exchanged through LDS each step.
__global__ __launch_bounds__(128)
void rnn_fused(const float* __restrict__ x,
               const float* __restrict__ W_ih,
               const float* __restrict__ W_hh,
               const float* __restrict__ b_ih,
               const float* __restrict__ b_hh,
               const float* __restrict__ W_fc,
               const float* __restrict__ b_fc,
               float* __restrict__ out)
{
    __shared__ __attribute__((aligned(16))) float hbuf[2 * HREG];

    const int tid  = threadIdx.x;
    const int lane = tid & 31;
    const int jt   = tid >> 5;         // wave id == output H-tile (rows 16jt..16jt+15)
    const int hi   = lane >> 4;        // lane half: K-pair select / M+8
    const int n    = lane & 15;        // A: row M ; B: col N ; D: col N
    const int b0   = blockIdx.x << 4;  // batch tile base

    // ---- this wave's W_hh A-layout (16x4 f32) tiles ----
    // A(M,K): lane = M + 16*(K>=2); VGPR v holds K = 2*hi + v (contiguous pair)
    v2f whh[16];
#pragma unroll
    for (int kk = 0; kk < 16; ++kk)
        whh[kk] = *(const v2f*)(W_hh + (jt*16 + n)*HDIM + kk*4 + hi*2);

    // ---- this wave's W_ih A-layout tiles (K over I=32) ----
    v2f wih[8];
#pragma unroll
    for (int kk = 0; kk < 8; ++kk)
        wih[kk] = *(const v2f*)(W_ih + (jt*16 + n)*IDIM + kk*4 + hi*2);

    // ---- bias (b_ih+b_hh) in D layout: VGPR r, lane-half -> row 16jt+r+8hi ----
    v8f biasD;
#pragma unroll
    for (int r = 0; r < 8; ++r) {
        const int j = jt*16 + r + hi*8;
        biasD[r] = b_ih[j] + b_hh[j];
    }

    v8f hsum = {};

    // h0 = 0: zero both LDS buffers
    for (int idx = tid; idx < 2*HREG; idx += 128) hbuf[idx] = 0.0f;
    __syncthreads();

    // per-lane base into x for B-layout chunks: lane loads
    // x[b0+n][tx][4kk + 2hi + {0,1}] as a float2
    const float* xlane = x + (size_t)(b0 + n) * TSEQ * IDIM + hi*2;

    // per-lane LDS offsets (dwords) in the pair-interleaved layout
    const int gbase = hi*PSTR + 2*n;                 // g chunk kk: + kk*2*PSTR
    const int sbase = (jt*8 + hi*4)*PSTR + 2*n;      // store pair r2: + r2*PSTR

    v2f bx[8];                          // software-pipelined x chunks for step t

    for (int t = 0; t < TP; ++t) {
        const int tx = t - PADL;        // uniform -> scalar branch

        // ---- x-projection chain (uses bx loaded during previous iteration) ----
        v8f accx;
        if (tx >= 0 && tx < TSEQ) {
            accx = __builtin_amdgcn_wmma_f32_16x16x4_f32(
                false, wih[0], false, bx[0], (short)0, biasD, false, false);
#pragma unroll
            for (int kk = 1; kk < 8; ++kk)
                accx = __builtin_amdgcn_wmma_f32_16x16x4_f32(
                    false, wih[kk], false, bx[kk], (short)0, accx, false, false);
        } else {
            accx = biasD;
        }

        // ---- issue next step's x loads; they complete under the recurrence ----
        {
            const int txn = tx + 1;
            if (txn >= 0 && txn < TSEQ) {
                const float* xp = xlane + (size_t)txn * IDIM;
                if (jt == 0 && txn + 16 < TSEQ)
                    __builtin_prefetch(xp + 16*IDIM, 0, 1);
#pragma unroll
                for (int kk = 0; kk < 8; ++kk)
                    bx[kk] = *(const v2f*)(xp + kk*4);
            }
        }

        // ---- recurrence: two independent 8-deep WMMA chains over hT chunks ----
        const float* hcur = hbuf + (t & 1) * HREG;
        const v8f z = {};
        v8f acch0 = __builtin_amdgcn_wmma_f32_16x16x4_f32(
            false, whh[0], false, *(const v2f*)(hcur + gbase), (short)0, z, false, false);
        v8f acch1 = __builtin_amdgcn_wmma_f32_16x16x4_f32(
            false, whh[8], false, *(const v2f*)(hcur + gbase + 8*2*PSTR), (short)0, z, false, false);
#pragma unroll
        for (int kk = 1; kk < 8; ++kk) {
            acch0 = __builtin_amdgcn_wmma_f32_16x16x4_f32(
                false, whh[kk], false, *(const v2f*)(hcur + gbase + kk*2*PSTR),
                (short)0, acch0, false, false);
            acch1 = __builtin_amdgcn_wmma_f32_16x16x4_f32(
                false, whh[kk + 8], false, *(const v2f*)(hcur + gbase + (kk + 8)*2*PSTR),
                (short)0, acch1, false, false);
        }
        const v8f pre = accx + acch0 + acch1;

        // ---- tanh (HW trans), publish h_new as b64 pairs, accumulate ----
        float* hnx = hbuf + ((t + 1) & 1) * HREG;
#pragma unroll
        for (int r2 = 0; r2 < 4; ++r2) {
            v2f hv;
            hv.x = hw_tanh(pre[2*r2 + 0]);
            hv.y = hw_tanh(pre[2*r2 + 1]);
            *(v2f*)(hnx + sbase + r2*PSTR) = hv;
            hsum[2*r2 + 0] += hv.x;
            hsum[2*r2 + 1] += hv.y;
        }
        __syncthreads();
    }

    // ---- FC head: out[b,o] = relu(b_fc[o] + sum_j (hsum[j,b]/Tp) * W_fc[o,j]) ----
    float* hs = hbuf;                   // reuse buffer 0 for hsumT (pair layout)
#pragma unroll
    for (int r2 = 0; r2 < 4; ++r2) {
        v2f hv; hv.x = hsum[2*r2]; hv.y = hsum[2*r2 + 1];
        *(v2f*)(hs + sbase + r2*PSTR) = hv;
    }
    __syncthreads();

    const float invTp = 1.0f / (float)TP;
    const int o  = tid & 15;            // output column
    const int br = tid >> 4;            // 0..7 -> batch rows {br, br+8}
#pragma unroll
    for (int half = 0; half < 2; ++half) {
        const int bc = br + half*8;
        float a = b_fc[o];
        for (int j = 0; j < HDIM; ++j)
            a = fmaf(W_fc[o*HDIM + j] * invTp, hs[(j >> 1)*PSTR + 2*bc + (j & 1)], a);
        out[(size_t)(b0 + bc)*ODIM + o] = fmaxf(a, 0.0f);
    }
}

extern "C" void kernel_launch(void* const* d_in, const int* in_sizes, int n_in,
                              void* d_out, int out_size, void* d_ws, size_t ws_size,
                              hipStream_t stream) {
    (void)in_sizes; (void)n_in; (void)out_size; (void)d_ws; (void)ws_size;
    const float* x    = (const float*)d_in[0];
    const float* W_ih = (const float*)d_in[1];
    const float* W_hh = (const float*)d_in[2];
    const float* b_ih = (const float*)d_in[3];
    const float* b_hh = (const float*)d_in[4];
    const float* W_fc = (const float*)d_in[5];
    const float* b_fc = (const float*)d_in[6];
    // B=256 -> 16 batch tiles; 4 waves per tile, one H-slice (jt) each
    rnn_fused<<<dim3(16), dim3(128), 0, stream>>>(x, W_ih, W_hh, b_ih, b_hh,
                                                  W_fc, b_fc, (float*)d_out);
}